// GRU_6579889898289
// MI455X (gfx1250) — compile-verified
//
#include <hip/hip_runtime.h>
#include <hip/hip_bf16.h>

// GRU on MI455X (gfx1250), fp32 WMMA (V_WMMA_F32_16X16X4_F32) throughout.
//
// Phases:
//  1) k_inproj : WX[g] = XS @ Wg^T + bg for g in {h,z,r}   (batched, parallel)
//  2) per timestep t (512 iterations, 2 kernels each):
//       k_step1: z = sigmoid(WXz[t] + h @ Uz^T)  -> Z
//                r = sigmoid(WXr[t] + h @ Ur^T)  -> RH = r*h
//       k_step2: hc = tanh(WXh[t] + RH @ Uh^T)
//                h  = z*h + (1-z)*hc ; ys[t] = h
//  3) k_outproj: out = tanh(ys @ Wl^T + bl)

#define T_STEPS 512
#define BATCH   32
#define IDIM    512
#define CDIM    1024
#define HDIM    1024
#define TB      (T_STEPS * BATCH)   // 16384

typedef float v2f __attribute__((ext_vector_type(2)));
typedef float v8f __attribute__((ext_vector_type(8)));

__device__ __forceinline__ v2f ld2(const float* __restrict__ p) {
    return *(const v2f*)p;
}

// D(16x16,f32) = A(16x4,f32) x B(4x16,f32) + C
__device__ __forceinline__ v8f wmma4(v2f a, v2f b, v8f c) {
    return __builtin_amdgcn_wmma_f32_16x16x4_f32(
        /*neg_a=*/false, a, /*neg_b=*/false, b,
        /*c_mod=*/(short)0, c, /*reuse_a=*/false, /*reuse_b=*/false);
}

__device__ __forceinline__ float sigmoidf_(float x) {
    return 1.0f / (1.0f + __expf(-x));
}

// ---------------------------------------------------------------------------
// Phase 1: input projections. One wave computes a 32x32 output tile (4 accs).
// grid = (CDIM/32, TB/32/8, 3), block = 256 (8 waves).
// ---------------------------------------------------------------------------
__global__ __launch_bounds__(256) void k_inproj(
    const float* __restrict__ X,                       // [TB, IDIM]
    const float* __restrict__ W0, const float* __restrict__ b0,   // h gate
    const float* __restrict__ W1, const float* __restrict__ b1,   // z gate
    const float* __restrict__ W2, const float* __restrict__ b2,   // r gate
    float* __restrict__ WX)                            // [3, TB, CDIM]
{
    const int lane = threadIdx.x & 31;
    const int wid  = threadIdx.x >> 5;
    const int half = lane >> 4;
    const int lr   = lane & 15;

    const int g = blockIdx.z;
    const float* Wg = (g == 0) ? W0 : ((g == 1) ? W1 : W2);
    const float* bg = (g == 0) ? b0 : ((g == 1) ? b1 : b2);

    const int n0 = blockIdx.x * 32;
    const int m0 = (blockIdx.y * 8 + wid) * 32;

    const float* A0 = X  + (size_t)(m0 + lr)      * IDIM;
    const float* A1 = X  + (size_t)(m0 + 16 + lr) * IDIM;
    const float* B0 = Wg + (size_t)(n0 + lr)      * IDIM;
    const float* B1 = Wg + (size_t)(n0 + 16 + lr) * IDIM;
    const int ko = 2 * half;

    v8f acc00 = {}, acc01 = {}, acc10 = {}, acc11 = {};
    for (int k = 0; k < IDIM; k += 4) {
        v2f a0 = ld2(A0 + k + ko);
        v2f a1 = ld2(A1 + k + ko);
        v2f bb0 = ld2(B0 + k + ko);
        v2f bb1 = ld2(B1 + k + ko);
        acc00 = wmma4(a0, bb0, acc00);
        acc01 = wmma4(a0, bb1, acc01);
        acc10 = wmma4(a1, bb0, acc10);
        acc11 = wmma4(a1, bb1, acc11);
    }

    float* out = WX + (size_t)g * TB * CDIM;
    const int n  = n0 + lr;
    const float bn0 = bg[n];
    const float bn1 = bg[n + 16];
#pragma unroll
    for (int j = 0; j < 8; ++j) {
        const int m = j + 8 * half;
        out[(size_t)(m0 + m)      * CDIM + n]      = acc00[j] + bn0;
        out[(size_t)(m0 + m)      * CDIM + n + 16] = acc01[j] + bn1;
        out[(size_t)(m0 + 16 + m) * CDIM + n]      = acc10[j] + bn0;
        out[(size_t)(m0 + 16 + m) * CDIM + n + 16] = acc11[j] + bn1;
    }
}

// ---------------------------------------------------------------------------
// Phase 2a: z and r gates for one timestep.
// 128 wave-tasks = 2 gates x 64 n-tiles. grid = (16), block = 256.
// Per wave: M=32 (full batch) x N=16, K split into two halves -> 4 indep accs.
// ---------------------------------------------------------------------------
__global__ __launch_bounds__(256) void k_step1(
    int t,
    const float* __restrict__ WX,    // [3, TB, CDIM]
    const float* __restrict__ Uz,    // [CDIM, CDIM]
    const float* __restrict__ Ur,    // [CDIM, CDIM]
    const float* __restrict__ h,     // [BATCH, CDIM]
    float* __restrict__ Z,           // [BATCH, CDIM]
    float* __restrict__ RH)          // [BATCH, CDIM]
{
    const int lane = threadIdx.x & 31;
    const int wid  = threadIdx.x >> 5;
    const int half = lane >> 4;
    const int lr   = lane & 15;

    const int task = blockIdx.x * 8 + wid;
    const int gate = task >> 6;            // 0 = z, 1 = r
    const int n0   = (task & 63) * 16;

    const float* U  = gate ? Ur : Uz;
    const float* wx = WX + (size_t)(gate ? 2 : 1) * TB * CDIM
                         + (size_t)t * BATCH * CDIM;

    const float* A0 = h + (size_t)lr        * CDIM;   // rows 0..15
    const float* A1 = h + (size_t)(16 + lr) * CDIM;   // rows 16..31
    const float* B  = U + (size_t)(n0 + lr) * CDIM;
    const int ko = 2 * half;
    const int KH = CDIM / 2;                          // 512

    v8f aLo0 = {}, aLo1 = {}, aHi0 = {}, aHi1 = {};
    for (int k = 0; k < KH; k += 4) {
        v2f x0 = ld2(A0 + k + ko);
        v2f x1 = ld2(A1 + k + ko);
        v2f x2 = ld2(A0 + KH + k + ko);
        v2f x3 = ld2(A1 + KH + k + ko);
        v2f bL = ld2(B + k + ko);
        v2f bH = ld2(B + KH + k + ko);
        aLo0 = wmma4(x0, bL, aLo0);
        aLo1 = wmma4(x1, bL, aLo1);
        aHi0 = wmma4(x2, bH, aHi0);
        aHi1 = wmma4(x3, bH, aHi1);
    }
    aLo0 += aHi0;   // combine K halves
    aLo1 += aHi1;

    const int n = n0 + lr;
#pragma unroll
    for (int j = 0; j < 8; ++j) {
        const int m0 = j + 8 * half;       // rows 0..15
        const int m1 = m0 + 16;            // rows 16..31
        float p0 = aLo0[j] + wx[(size_t)m0 * CDIM + n];
        float p1 = aLo1[j] + wx[(size_t)m1 * CDIM + n];
        float s0 = sigmoidf_(p0);
        float s1 = sigmoidf_(p1);
        if (gate) {
            RH[(size_t)m0 * CDIM + n] = s0 * h[(size_t)m0 * CDIM + n];
            RH[(size_t)m1 * CDIM + n] = s1 * h[(size_t)m1 * CDIM + n];
        } else {
            Z[(size_t)m0 * CDIM + n] = s0;
            Z[(size_t)m1 * CDIM + n] = s1;
        }
    }
}

// ---------------------------------------------------------------------------
// Phase 2b: candidate + blend for one timestep.
// 64 wave-tasks. grid = (8), block = 256.
// ---------------------------------------------------------------------------
__global__ __launch_bounds__(256) void k_step2(
    int t,
    const float* __restrict__ WX,    // [3, TB, CDIM] (gate 0 = h)
    const float* __restrict__ Uh,    // [CDIM, CDIM]
    const float* __restrict__ RH,    // [BATCH, CDIM]
    const float* __restrict__ Z,     // [BATCH, CDIM]
    float* __restrict__ h,           // [BATCH, CDIM] (read + write)
    float* __restrict__ ys)          // [TB, CDIM]
{
    const int lane = threadIdx.x & 31;
    const int wid  = threadIdx.x >> 5;
    const int half = lane >> 4;
    const int lr   = lane & 15;

    const int n0 = (blockIdx.x * 8 + wid) * 16;

    const float* wx = WX + (size_t)t * BATCH * CDIM;  // h-gate slab
    const float* A0 = RH + (size_t)lr        * CDIM;
    const float* A1 = RH + (size_t)(16 + lr) * CDIM;
    const float* B  = Uh + (size_t)(n0 + lr) * CDIM;
    const int ko = 2 * half;
    const int KH = CDIM / 2;

    v8f aLo0 = {}, aLo1 = {}, aHi0 = {}, aHi1 = {};
    for (int k = 0; k < KH; k += 4) {
        v2f x0 = ld2(A0 + k + ko);
        v2f x1 = ld2(A1 + k + ko);
        v2f x2 = ld2(A0 + KH + k + ko);
        v2f x3 = ld2(A1 + KH + k + ko);
        v2f bL = ld2(B + k + ko);
        v2f bH = ld2(B + KH + k + ko);
        aLo0 = wmma4(x0, bL, aLo0);
        aLo1 = wmma4(x1, bL, aLo1);
        aHi0 = wmma4(x2, bH, aHi0);
        aHi1 = wmma4(x3, bH, aHi1);
    }
    aLo0 += aHi0;
    aLo1 += aHi1;

    const int n = n0 + lr;
    float* yrow = ys + (size_t)t * BATCH * CDIM;
#pragma unroll
    for (int j = 0; j < 8; ++j) {
        const int m0 = j + 8 * half;
        const int m1 = m0 + 16;

        float hc0 = tanhf(aLo0[j] + wx[(size_t)m0 * CDIM + n]);
        float hc1 = tanhf(aLo1[j] + wx[(size_t)m1 * CDIM + n]);
        float z0 = Z[(size_t)m0 * CDIM + n];
        float z1 = Z[(size_t)m1 * CDIM + n];
        float hp0 = h[(size_t)m0 * CDIM + n];
        float hp1 = h[(size_t)m1 * CDIM + n];
        float hn0 = z0 * hp0 + (1.0f - z0) * hc0;
        float hn1 = z1 * hp1 + (1.0f - z1) * hc1;
        h[(size_t)m0 * CDIM + n] = hn0;
        h[(size_t)m1 * CDIM + n] = hn1;
        yrow[(size_t)m0 * CDIM + n] = hn0;
        yrow[(size_t)m1 * CDIM + n] = hn1;
    }
}

// ---------------------------------------------------------------------------
// Phase 3: output projection. out = tanh(ys @ Wl^T + bl).
// grid = (HDIM/32, TB/32/8), block = 256.
// ---------------------------------------------------------------------------
__global__ __launch_bounds__(256) void k_outproj(
    const float* __restrict__ ys,    // [TB, CDIM]
    const float* __restrict__ Wl,    // [HDIM, CDIM]
    const float* __restrict__ bl,    // [HDIM]
    float* __restrict__ out)         // [TB, HDIM]
{
    const int lane = threadIdx.x & 31;
    const int wid  = threadIdx.x >> 5;
    const int half = lane >> 4;
    const int lr   = lane & 15;

    const int n0 = blockIdx.x * 32;
    const int m0 = (blockIdx.y * 8 + wid) * 32;

    const float* A0 = ys + (size_t)(m0 + lr)      * CDIM;
    const float* A1 = ys + (size_t)(m0 + 16 + lr) * CDIM;
    const float* B0 = Wl + (size_t)(n0 + lr)      * CDIM;
    const float* B1 = Wl + (size_t)(n0 + 16 + lr) * CDIM;
    const int ko = 2 * half;

    v8f acc00 = {}, acc01 = {}, acc10 = {}, acc11 = {};
    for (int k = 0; k < CDIM; k += 4) {
        v2f a0 = ld2(A0 + k + ko);
        v2f a1 = ld2(A1 + k + ko);
        v2f bb0 = ld2(B0 + k + ko);
        v2f bb1 = ld2(B1 + k + ko);
        acc00 = wmma4(a0, bb0, acc00);
        acc01 = wmma4(a0, bb1, acc01);
        acc10 = wmma4(a1, bb0, acc10);
        acc11 = wmma4(a1, bb1, acc11);
    }

    const int n = n0 + lr;
    const float bn0 = bl[n];
    const float bn1 = bl[n + 16];
#pragma unroll
    for (int j = 0; j < 8; ++j) {
        const int m = j + 8 * half;
        out[(size_t)(m0 + m)      * HDIM + n]      = tanhf(acc00[j] + bn0);
        out[(size_t)(m0 + m)      * HDIM + n + 16] = tanhf(acc01[j] + bn1);
        out[(size_t)(m0 + 16 + m) * HDIM + n]      = tanhf(acc10[j] + bn0);
        out[(size_t)(m0 + 16 + m) * HDIM + n + 16] = tanhf(acc11[j] + bn1);
    }
}

// ---------------------------------------------------------------------------
extern "C" void kernel_launch(void* const* d_in, const int* in_sizes, int n_in,
                              void* d_out, int out_size, void* d_ws, size_t ws_size,
                              hipStream_t stream) {
    const float* xs = (const float*)d_in[0];
    // d_in[1] = ilens : unused by the reference computation
    const float* Wh = (const float*)d_in[2];
    const float* bh = (const float*)d_in[3];
    const float* Uh = (const float*)d_in[4];
    const float* Wz = (const float*)d_in[5];
    const float* bz = (const float*)d_in[6];
    const float* Uz = (const float*)d_in[7];
    const float* Wr = (const float*)d_in[8];
    const float* br = (const float*)d_in[9];
    const float* Ur = (const float*)d_in[10];
    const float* Wl = (const float*)d_in[11];
    const float* bl = (const float*)d_in[12];

    // Workspace layout (floats):
    //   WX : 3 * TB * CDIM        = 50,331,648
    //   ys : TB * CDIM            = 16,777,216
    //   h  : BATCH * CDIM         = 32,768
    //   Z  : BATCH * CDIM         = 32,768
    //   RH : BATCH * CDIM         = 32,768
    float* ws = (float*)d_ws;
    float* WX = ws;
    float* ys = WX + (size_t)3 * TB * CDIM;
    float* h  = ys + (size_t)TB * CDIM;
    float* Z  = h  + (size_t)BATCH * CDIM;
    float* RH = Z  + (size_t)BATCH * CDIM;
    (void)ws_size; (void)in_sizes; (void)n_in; (void)out_size;

    // h0 = 0
    hipMemsetAsync(h, 0, (size_t)BATCH * CDIM * sizeof(float), stream);

    // Phase 1: batched input projections for all timesteps
    k_inproj<<<dim3(CDIM / 32, TB / 32 / 8, 3), 256, 0, stream>>>(
        xs, Wh, bh, Wz, bz, Wr, br, WX);

    // Phase 2: sequential recurrence
    for (int t = 0; t < T_STEPS; ++t) {
        k_step1<<<dim3(16), 256, 0, stream>>>(t, WX, Uz, Ur, h, Z, RH);
        k_step2<<<dim3(8),  256, 0, stream>>>(t, WX, Uh, RH, Z, h, ys);
    }

    // Phase 3: output projection
    k_outproj<<<dim3(HDIM / 32, TB / 32 / 8), 256, 0, stream>>>(
        ys, Wl, bl, (float*)d_out);
}